// myGNN_17016660427424
// MI455X (gfx1250) — compile-verified
//
#include <hip/hip_runtime.h>
#include <hip/hip_bf16.h>
#include <math.h>

typedef float v2f __attribute__((ext_vector_type(2)));
typedef float v8f __attribute__((ext_vector_type(8)));

#define F 16

// ---------------------------------------------------------------------------
// Zero a float buffer (grid-stride).
__global__ void zero_f32(float* __restrict__ p, size_t n) {
    size_t i = (size_t)blockIdx.x * blockDim.x + threadIdx.x;
    size_t stride = (size_t)gridDim.x * blockDim.x;
    for (; i < n; i += stride) p[i] = 0.0f;
}

// ---------------------------------------------------------------------------
// One-time: int64 edge indices -> int32, and accumulate in-degree (float).
__global__ void convert_edges(const long long* __restrict__ src64,
                              const long long* __restrict__ dst64,
                              int* __restrict__ src32, int* __restrict__ dst32,
                              float* __restrict__ deg, int E) {
    int e = blockIdx.x * blockDim.x + threadIdx.x;
    if (e >= E) return;
    int s = (int)src64[e];
    int d = (int)dst64[e];
    src32[e] = s;
    dst32[e] = d;
    __hip_atomic_fetch_add(&deg[d], 1.0f, __ATOMIC_RELAXED, __HIP_MEMORY_SCOPE_AGENT);
}

// Fallback (small workspace): degree only, reading int64 directly.
__global__ void count_deg64(const long long* __restrict__ dst64,
                            float* __restrict__ deg, int E) {
    int e = blockIdx.x * blockDim.x + threadIdx.x;
    if (e >= E) return;
    __hip_atomic_fetch_add(&deg[(int)dst64[e]], 1.0f, __ATOMIC_RELAXED, __HIP_MEMORY_SCOPE_AGENT);
}

// deg -> 1/max(deg,1) in place.
__global__ void invert_deg(float* __restrict__ deg, int N) {
    int i = blockIdx.x * blockDim.x + threadIdx.x;
    if (i >= N) return;
    deg[i] = 1.0f / fmaxf(deg[i], 1.0f);
}

// ---------------------------------------------------------------------------
// Edge scatter: agg[dst] += x[src]  (16 fp32 lanes per edge, native f32 atomics
// into the L2-resident agg table). Templated on index type for the ws fallback.
template <typename IdxT>
__global__ void scatter_edges(const IdxT* __restrict__ src, const IdxT* __restrict__ dst,
                              const float* __restrict__ x, float* __restrict__ agg, int E) {
    int e = blockIdx.x * blockDim.x + threadIdx.x;
    if (e >= E) return;
    int s = (int)src[e];
    int d = (int)dst[e];
    const float4* xr = (const float4*)(x + (size_t)s * F);
    float4 r0 = xr[0], r1 = xr[1], r2 = xr[2], r3 = xr[3];
    float* a = agg + (size_t)d * F;
#define ATOM(off, v) __hip_atomic_fetch_add(a + (off), (v), __ATOMIC_RELAXED, __HIP_MEMORY_SCOPE_AGENT)
    ATOM(0, r0.x);  ATOM(1, r0.y);  ATOM(2, r0.z);  ATOM(3, r0.w);
    ATOM(4, r1.x);  ATOM(5, r1.y);  ATOM(6, r1.z);  ATOM(7, r1.w);
    ATOM(8, r2.x);  ATOM(9, r2.y);  ATOM(10, r2.z); ATOM(11, r2.w);
    ATOM(12, r3.x); ATOM(13, r3.y); ATOM(14, r3.z); ATOM(15, r3.w);
#undef ATOM
}

// ---------------------------------------------------------------------------
// Per-node transform via V_WMMA_F32_16X16X4_F32. One wave32 handles one
// 16-node tile. Two fused GEMMs:
//   out = relu( (agg*invdeg) @ Wl^T + bl + xin @ Wr^T )     (agg != null)
//   out = relu( xin @ Wl^T + bl )                           (agg == null, fc1)
// Requires N % 16 == 0 (true here: 200000 / 16 = 12500).
__global__ void sage_transform(const float* __restrict__ xin,
                               const float* __restrict__ agg,
                               const float* __restrict__ invdeg,
                               const float* __restrict__ Wl,
                               const float* __restrict__ bl,
                               const float* __restrict__ Wr,
                               float* __restrict__ out,
                               int nTiles) {
    const int lane = threadIdx.x & 31;
    const int wave = (int)((blockIdx.x * blockDim.x + threadIdx.x) >> 5);
    if (wave >= nTiles) return;  // wave-uniform: EXEC stays all-1s past here

    const int nb   = wave << 4;     // first node of the tile
    const int half = lane >> 4;     // 0: K=k0..k0+1, 1: K=k0+2..k0+3
    const int l16  = lane & 15;     // M (for A) / N (for B, C, D)
    const int row  = nb + l16;

    // C/D init: broadcast bias over M (VGPR index), per-N (lane) value.
    const float bias = bl[l16];
    v8f acc;
#pragma unroll
    for (int i = 0; i < 8; ++i) acc[i] = bias;

    if (agg != nullptr) {
        const float inv = invdeg[row];
        // GEMM 1: mean @ Wl^T, K split into 4 chunks of 4.
#pragma unroll
        for (int c = 0; c < 4; ++c) {
            const int k0 = (c << 2) + (half << 1);
            v2f a, b;
            a[0] = agg[row * F + k0]     * inv;
            a[1] = agg[row * F + k0 + 1] * inv;
            b[0] = Wl[l16 * F + k0];        // B[k][n] = Wl[n][k]
            b[1] = Wl[l16 * F + k0 + 1];
            acc = __builtin_amdgcn_wmma_f32_16x16x4_f32(false, a, false, b,
                                                        (short)0, acc, false, false);
        }
        // GEMM 2: xin @ Wr^T, accumulate into same tile.
#pragma unroll
        for (int c = 0; c < 4; ++c) {
            const int k0 = (c << 2) + (half << 1);
            v2f a, b;
            a[0] = xin[row * F + k0];
            a[1] = xin[row * F + k0 + 1];
            b[0] = Wr[l16 * F + k0];
            b[1] = Wr[l16 * F + k0 + 1];
            acc = __builtin_amdgcn_wmma_f32_16x16x4_f32(false, a, false, b,
                                                        (short)0, acc, false, false);
        }
    } else {
        // fc1 path: single GEMM xin @ Wl^T.
#pragma unroll
        for (int c = 0; c < 4; ++c) {
            const int k0 = (c << 2) + (half << 1);
            v2f a, b;
            a[0] = xin[row * F + k0];
            a[1] = xin[row * F + k0 + 1];
            b[0] = Wl[l16 * F + k0];
            b[1] = Wl[l16 * F + k0 + 1];
            acc = __builtin_amdgcn_wmma_f32_16x16x4_f32(false, a, false, b,
                                                        (short)0, acc, false, false);
        }
    }

    // Store with ReLU. D element (M = v + 8*half, N = l16).
#pragma unroll
    for (int v = 0; v < 8; ++v) {
        float val = fmaxf(acc[v], 0.0f);
        out[(size_t)(nb + v + (half << 3)) * F + l16] = val;
    }
}

// ---------------------------------------------------------------------------
// Final head: slice h[:, :8], 8x8 FC + bias, ReLU, softmax over 8 classes.
__global__ void fc2_softmax(const float* __restrict__ h, const float* __restrict__ W,
                            const float* __restrict__ b, float* __restrict__ out, int N) {
    __shared__ float sW[64];
    __shared__ float sb[8];
    if (threadIdx.x < 64) sW[threadIdx.x] = W[threadIdx.x];
    if (threadIdx.x < 8)  sb[threadIdx.x] = b[threadIdx.x];
    __syncthreads();
    int i = blockIdx.x * blockDim.x + threadIdx.x;
    if (i >= N) return;

    const float4* hp = (const float4*)(h + (size_t)i * F);
    float4 h0 = hp[0], h1 = hp[1];
    float hin[8] = {h0.x, h0.y, h0.z, h0.w, h1.x, h1.y, h1.z, h1.w};

    float y[8];
    float m = -1e30f;
#pragma unroll
    for (int n = 0; n < 8; ++n) {
        float acc = sb[n];
#pragma unroll
        for (int k = 0; k < 8; ++k) acc = fmaf(hin[k], sW[n * 8 + k], acc);
        acc = fmaxf(acc, 0.0f);
        y[n] = acc;
        m = fmaxf(m, acc);
    }
    float s = 0.0f;
#pragma unroll
    for (int n = 0; n < 8; ++n) { float e = __expf(y[n] - m); y[n] = e; s += e; }
    float r = 1.0f / s;
    float* o = out + (size_t)i * 8;
#pragma unroll
    for (int n = 0; n < 8; ++n) o[n] = y[n] * r;
}

// ---------------------------------------------------------------------------
extern "C" void kernel_launch(void* const* d_in, const int* in_sizes, int n_in,
                              void* d_out, int out_size, void* d_ws, size_t ws_size,
                              hipStream_t stream) {
    const float*     x      = (const float*)d_in[0];
    const long long* ei     = (const long long*)d_in[1];  // (2, E) int64
    const float*     c1_Wl  = (const float*)d_in[2];      // (2,16,16)
    const float*     c1_bl  = (const float*)d_in[3];      // (2,16)
    const float*     c1_Wr  = (const float*)d_in[4];
    const float*     c2_Wl  = (const float*)d_in[5];
    const float*     c2_bl  = (const float*)d_in[6];
    const float*     c2_Wr  = (const float*)d_in[7];
    const float*     fc1_W  = (const float*)d_in[8];
    const float*     fc1_b  = (const float*)d_in[9];
    const float*     fc2_W  = (const float*)d_in[10];
    const float*     fc2_b  = (const float*)d_in[11];

    const int N = in_sizes[0] / F;       // 200000
    const int E = in_sizes[1] / 2;       // 5000000
    const long long* src64 = ei;
    const long long* dst64 = ei + E;

    // Workspace layout (all offsets 16B-aligned for given N, E).
    char* w = (char*)d_ws;
    size_t offDeg = 0;                                   // N floats (deg -> invdeg)
    size_t offAgg = offDeg + (size_t)N * 4;              // N*16 floats
    size_t offHA  = offAgg + (size_t)N * F * 4;          // N*16 floats
    size_t offHB  = offHA  + (size_t)N * F * 4;          // N*16 floats
    size_t offSrc = offHB  + (size_t)N * F * 4;          // E ints (optional)
    size_t offDst = offSrc + (size_t)E * 4;
    size_t needBig = offDst + (size_t)E * 4;

    float* deg    = (float*)(w + offDeg);   // becomes invdeg in place
    float* agg    = (float*)(w + offAgg);
    float* hA     = (float*)(w + offHA);
    float* hB     = (float*)(w + offHB);
    int*   src32  = (int*)(w + offSrc);
    int*   dst32  = (int*)(w + offDst);
    const bool use32 = (ws_size >= needBig);   // host param -> deterministic

    const int TB = 256;
    const int edgeBlocks = (E + TB - 1) / TB;
    const int nodeBlocks = (N + TB - 1) / TB;
    const int nTiles = N / F;                              // 12500 (N % 16 == 0)
    const int wavesPerBlock = TB / 32;
    const int tfmBlocks = (nTiles + wavesPerBlock - 1) / wavesPerBlock;
    const int zeroBlocksAgg = 2048;

    // --- One-time structure prep: deg + (optionally) int32 edge copy -------
    zero_f32<<<nodeBlocks, TB, 0, stream>>>(deg, (size_t)N);
    if (use32) {
        convert_edges<<<edgeBlocks, TB, 0, stream>>>(src64, dst64, src32, dst32, deg, E);
    } else {
        count_deg64<<<edgeBlocks, TB, 0, stream>>>(dst64, deg, E);
    }
    invert_deg<<<nodeBlocks, TB, 0, stream>>>(deg, N);

    // --- One SAGEConv pass -------------------------------------------------
    auto conv = [&](const float* xin, float* xout, const float* Wl,
                    const float* blp, const float* Wr) {
        zero_f32<<<zeroBlocksAgg, TB, 0, stream>>>(agg, (size_t)N * F);
        if (use32) {
            scatter_edges<int><<<edgeBlocks, TB, 0, stream>>>(src32, dst32, xin, agg, E);
        } else {
            scatter_edges<long long><<<edgeBlocks, TB, 0, stream>>>(src64, dst64, xin, agg, E);
        }
        sage_transform<<<tfmBlocks, TB, 0, stream>>>(xin, agg, deg, Wl, blp, Wr, xout, nTiles);
    };

    // GraphSAGE block 1 (relu between layers AND relu on output -> always relu)
    conv(x,  hA, c1_Wl,       c1_bl,      c1_Wr);
    conv(hA, hB, c1_Wl + 256, c1_bl + 16, c1_Wr + 256);
    // fc1 (relu)
    sage_transform<<<tfmBlocks, TB, 0, stream>>>(hB, nullptr, nullptr,
                                                 fc1_W, fc1_b, nullptr, hA, nTiles);
    // GraphSAGE block 2
    conv(hA, hB, c2_Wl,       c2_bl,      c2_Wr);
    conv(hB, hA, c2_Wl + 256, c2_bl + 16, c2_Wr + 256);
    // slice[:, :8] + fc2 + relu + softmax
    fc2_softmax<<<nodeBlocks, TB, 0, stream>>>(hA, fc2_W, fc2_b, (float*)d_out, N);
}